// MyModel_61933428416005
// MI455X (gfx1250) — compile-verified
//
#include <hip/hip_runtime.h>

// MI455X / gfx1250, wave32. Bandwidth-bound (~272 MB @ 23.3 TB/s ~= 12 us floor).
// Per-track 4x4 Cholesky + triangular inverse: thread-per-track into LDS (amortized).
// Projection z = Linv*(x-mu): V_WMMA_F32_16X16X4_F32, oriented so ||z||^2 reduces
// within a single lane. Streaming loop batches all 4 tile loads before the WMMAs
// so the HBM pipe sees 1KB of requests per track at once.

typedef float v2f __attribute__((ext_vector_type(2)));
typedef float v8f __attribute__((ext_vector_type(8)));

__global__ __launch_bounds__(256) void gate_dist_kernel(
    const float* __restrict__ mean,     // [N,4]
    const float* __restrict__ cov,      // [N,4,4]
    const float* __restrict__ meas,     // [N,64,4]
    const float* __restrict__ umat,     // [4,4]
    float* __restrict__ out,            // [N,64]
    int N)
{
    // 256 tracks/block, 20 floats per packet: Linv (4x4 row-major, upper=0) + new_mean (4)
    __shared__ alignas(16) float pkt[256 * 20];

    const int lt    = threadIdx.x;
    const int base  = blockIdx.x << 8;
    const int track = base + lt;

    if (track < N) {
        // ---- uniform update matrix ----
        float U[16];
        #pragma unroll
        for (int i = 0; i < 4; ++i) {
            float4 u = reinterpret_cast<const float4*>(umat)[i];
            U[i*4+0] = u.x; U[i*4+1] = u.y; U[i*4+2] = u.z; U[i*4+3] = u.w;
        }
        // ---- per-track mean & covariance ----
        float4 m4 = reinterpret_cast<const float4*>(mean)[track];
        float m[4] = {m4.x, m4.y, m4.z, m4.w};
        float C[16];
        #pragma unroll
        for (int i = 0; i < 4; ++i) {
            float4 c4 = reinterpret_cast<const float4*>(cov)[track*4 + i];
            C[i*4+0] = c4.x; C[i*4+1] = c4.y; C[i*4+2] = c4.z; C[i*4+3] = c4.w;
        }
        // new_mean = mean @ U^T
        float nm[4];
        #pragma unroll
        for (int i = 0; i < 4; ++i)
            nm[i] = U[i*4+0]*m[0] + U[i*4+1]*m[1] + U[i*4+2]*m[2] + U[i*4+3]*m[3];
        // T = C @ U^T ; S = U @ T  (lower triangle only)
        float T[16];
        #pragma unroll
        for (int j = 0; j < 4; ++j)
            #pragma unroll
            for (int l = 0; l < 4; ++l)
                T[j*4+l] = C[j*4+0]*U[l*4+0] + C[j*4+1]*U[l*4+1]
                         + C[j*4+2]*U[l*4+2] + C[j*4+3]*U[l*4+3];
        float S[4][4];
        #pragma unroll
        for (int i = 0; i < 4; ++i)
            #pragma unroll
            for (int l = 0; l <= i; ++l)
                S[i][l] = U[i*4+0]*T[0*4+l] + U[i*4+1]*T[1*4+l]
                        + U[i*4+2]*T[2*4+l] + U[i*4+3]*T[3*4+l];
        // innovation diag: std = [0.1h, 0.1h, 0.1, 0.1h], h = mean[3]
        const float sw  = 0.1f * m[3];
        const float sw2 = sw * sw;
        S[0][0] += sw2; S[1][1] += sw2; S[2][2] += 0.01f; S[3][3] += sw2;

        // ---- Cholesky (lower) ----
        float L00 = sqrtf(S[0][0]);
        float r0  = 1.0f / L00;
        float L10 = S[1][0]*r0, L20 = S[2][0]*r0, L30 = S[3][0]*r0;
        float L11 = sqrtf(S[1][1] - L10*L10);
        float r1  = 1.0f / L11;
        float L21 = (S[2][1] - L20*L10)*r1;
        float L31 = (S[3][1] - L30*L10)*r1;
        float L22 = sqrtf(S[2][2] - L20*L20 - L21*L21);
        float r2  = 1.0f / L22;
        float L32 = (S[3][2] - L30*L20 - L31*L21)*r2;
        float L33 = sqrtf(S[3][3] - L30*L30 - L31*L31 - L32*L32);
        float r3  = 1.0f / L33;

        // ---- explicit triangular inverse Linv = L^-1 ----
        float Li10 = -r1 * (L10*r0);
        float Li20 = -r2 * (L20*r0 + L21*Li10);
        float Li21 = -r2 * (L21*r1);
        float Li30 = -r3 * (L30*r0 + L31*Li10 + L32*Li20);
        float Li31 = -r3 * (L31*r1 + L32*Li21);
        float Li32 = -r3 * (L32*r2);

        float* p = &pkt[lt * 20];
        reinterpret_cast<float4*>(p)[0] = make_float4(r0,   0.f,  0.f,  0.f);
        reinterpret_cast<float4*>(p)[1] = make_float4(Li10, r1,   0.f,  0.f);
        reinterpret_cast<float4*>(p)[2] = make_float4(Li20, Li21, r2,   0.f);
        reinterpret_cast<float4*>(p)[3] = make_float4(Li30, Li31, Li32, r3);
        reinterpret_cast<float4*>(p)[4] = make_float4(nm[0], nm[1], nm[2], nm[3]);
    }
    __syncthreads();

    // ---- streaming phase: each wave handles 32 tracks ----
    const int wave = lt >> 5;
    const int lane = lt & 31;
    const int mrow = lane & 15;   // A row / B column index
    const int kh   = lane >> 4;   // which K-pair (k = 2*kh, 2*kh+1)

    for (int it = 0; it < 32; ++it) {
        const int ltr = (wave << 5) | it;
        const int tr  = base + ltr;             // wave-uniform
        if (tr >= N) break;                     // uniform branch: EXEC stays full

        const float* p = &pkt[ltr * 20];
        // A = Linv rows 0..3 (rows 4..15 zero): lane holds Linv[mrow][2kh..2kh+1]
        v2f a;
        if (mrow < 4) a = *reinterpret_cast<const v2f*>(p + mrow*4 + kh*2);
        else { a.x = 0.f; a.y = 0.f; }
        const v2f muv = *reinterpret_cast<const v2f*>(p + 16 + kh*2);

        // lane reads components (2kh,2kh+1) of measurement (tile*16 + mrow):
        // bytes tr*1024 + mrow*16 + kh*8 -> wave covers 256 contiguous bytes/tile
        const float* mp = meas + (size_t)tr*256 + (size_t)(mrow*4 + kh*2);
        float* op = out + (size_t)tr*64 + mrow;

        // --- issue all 4 tile loads back-to-back (1KB in flight per wave) ---
        v2f mv[4];
        #pragma unroll
        for (int t = 0; t < 4; ++t)
            mv[t] = __builtin_nontemporal_load(
                        reinterpret_cast<const v2f*>(mp + t*64));

        // --- 4 WMMAs + per-lane reductions ---
        float s[4];
        #pragma unroll
        for (int t = 0; t < 4; ++t) {
            v2f b;  // B[k][n] = meas[n][k] - mu[k]
            b.x = mv[t].x - muv.x;
            b.y = mv[t].y - muv.y;
            v8f c = {0.f,0.f,0.f,0.f,0.f,0.f,0.f,0.f};
            // D[i][n] = sum_k Linv[i][k] * d[n][k] = z_i of measurement n
            v8f d = __builtin_amdgcn_wmma_f32_16x16x4_f32(
                        false, a, false, b, (short)0, c, false, false);
            // lane n (0..15) holds z0..z3 in d[0..3]; rows 4..7 are zero
            s[t] = d[0]*d[0] + d[1]*d[1] + d[2]*d[2] + d[3]*d[3];
        }

        // --- single predicated store block: 4 NT b32 stores, imm offsets ---
        if (lane < 16) {
            #pragma unroll
            for (int t = 0; t < 4; ++t)
                __builtin_nontemporal_store(s[t], op + t*16);
        }
    }
}

extern "C" void kernel_launch(void* const* d_in, const int* in_sizes, int n_in,
                              void* d_out, int out_size, void* d_ws, size_t ws_size,
                              hipStream_t stream) {
    (void)n_in; (void)out_size; (void)d_ws; (void)ws_size;
    const float* mean = (const float*)d_in[0];
    const float* cov  = (const float*)d_in[1];
    const float* meas = (const float*)d_in[2];
    const float* umat = (const float*)d_in[3];
    float* out = (float*)d_out;
    const int N = in_sizes[0] / 4;          // mean is [N,4]
    const int blocks = (N + 255) / 256;
    gate_dist_kernel<<<blocks, 256, 0, stream>>>(mean, cov, meas, umat, out, N);
}